// RGCN_Net_52965536694389
// MI455X (gfx1250) — compile-verified
//
#include <hip/hip_runtime.h>
#include <hip/hip_bf16.h>

// ---------------------------------------------------------------------------
// RGCN link-prediction forward for MI455X (gfx1250, wave32).
//
// Sizes: N=15546 nodes, IN=3000, H1=64, OUT=4, R=3, NB=1, E=100000, ED=50000.
//
// NB=1 lets us fold comp[r]/cnt[n,r] into the edge-scatter coefficient:
//   out = (sum_r comp[r]*mean_r) @ B  + x @ root + bias
// so message passing needs only one [N,3000] accumulator and ONE big GEMM
// pair: [15546,3000] x [3000,64], done with V_WMMA_F32_16X16X4_F32.
//
// Roofline: ~1.2 GB gathered reads + ~1.2 GB GLOBAL_ATOMIC_ADD_F32 scatter
// dominates (~150 us @ 23.3 TB/s); the 12 GFLOP f32 GEMM rides on WMMA.
// ---------------------------------------------------------------------------

typedef __attribute__((ext_vector_type(2))) float v2f;
typedef __attribute__((ext_vector_type(8))) float v8f;

#define NNODES 15546
#define INFEAT 3000
#define H1DIM  64
#define OUTDIM 4
#define NREL   3
#define NEDGE  100000
#define EDEC   50000

// ---- edge counts per (tgt, relation): shared by both RGCN layers ----------
__global__ void count_edges(const int* __restrict__ ei,
                            const int* __restrict__ et,
                            int* __restrict__ cnt) {
    int e = blockIdx.x * blockDim.x + threadIdx.x;
    if (e < NEDGE) {
        int tgt = ei[NEDGE + e];
        int r   = et[e];
        atomicAdd(&cnt[tgt * NREL + r], 1);
    }
}

// ---- layer-1 scatter: agg1[tgt,:] += (comp1[r]/cnt[tgt,r]) * x[src,:] -----
// one block per edge, 256 threads stride over the 3000-wide row (coalesced).
__global__ __launch_bounds__(256) void scatter1(
    const float* __restrict__ x, const int* __restrict__ ei,
    const int* __restrict__ et, const float* __restrict__ comp1,
    const int* __restrict__ cnt, float* __restrict__ agg1) {
    int e   = blockIdx.x;
    int src = ei[e];
    int tgt = ei[NEDGE + e];
    int r   = et[e];
    int c   = cnt[tgt * NREL + r];
    float coef = comp1[r] / (float)(c > 1 ? c : 1);
    const float* xr = x    + (size_t)src * INFEAT;
    float*       ar = agg1 + (size_t)tgt * INFEAT;
    for (int j = threadIdx.x; j < INFEAT; j += 256)
        unsafeAtomicAdd(&ar[j], coef * xr[j]);   // GLOBAL_ATOMIC_ADD_F32
}

// ---- GEMM1 + bias + ReLU via V_WMMA_F32_16X16X4_F32 -----------------------
// h = relu(agg1 @ B1 + x @ root1 + bias1).  One wave per 16-row strip,
// 4 independent 16x16 accumulator tiles covering all 64 output columns.
__global__ __launch_bounds__(32) void gemm1_wmma_relu(
    const float* __restrict__ agg1, const float* __restrict__ x,
    const float* __restrict__ B1,   /* bases1[0]: [3000,64] */
    const float* __restrict__ root1,/* [3000,64] */
    const float* __restrict__ bias1,/* [64] */
    float* __restrict__ h) {
    const int m0   = blockIdx.x * 16;
    const int lane = threadIdx.x;      // 0..31
    const int half = lane >> 4;        // K-pair select (ISA A/B layout)
    const int m    = lane & 15;        // M for A loads / N for B loads

    int rowA = m0 + m;
    int rowC = rowA < NNODES ? rowA : 0;   // clamp OOB rows (store is masked)
    const float* aR = agg1 + (size_t)rowC * INFEAT;
    const float* xR = x    + (size_t)rowC * INFEAT;

    v8f acc[4] = {};
#pragma unroll 2
    for (int k0 = 0; k0 < INFEAT; k0 += 4) {
        int ka = k0 + 2 * half;                        // this lane's K base
        v2f aA = { aR[ka], aR[ka + 1] };               // A tile of agg1
        v2f aX = { xR[ka], xR[ka + 1] };               // A tile of x
#pragma unroll
        for (int t = 0; t < 4; ++t) {
            int nb = t * 16;
            v2f b1 = { B1[(size_t)ka * H1DIM + nb + m],
                       B1[(size_t)(ka + 1) * H1DIM + nb + m] };
            v2f br = { root1[(size_t)ka * H1DIM + nb + m],
                       root1[(size_t)(ka + 1) * H1DIM + nb + m] };
            acc[t] = __builtin_amdgcn_wmma_f32_16x16x4_f32(
                false, aA, false, b1, (short)0, acc[t], false, false);
            acc[t] = __builtin_amdgcn_wmma_f32_16x16x4_f32(
                false, aX, false, br, (short)0, acc[t], false, false);
        }
    }

    // C/D layout: VGPR g -> row m0 + g + 8*half, col nb + m.
    // 971/972 tiles are full -> uniform branch, unguarded coalesced stores.
    const bool full = (m0 + 16) <= NNODES;
#pragma unroll
    for (int t = 0; t < 4; ++t) {
        int nb = t * 16;
        float bb = bias1[nb + m];
        if (full) {
#pragma unroll
            for (int g = 0; g < 8; ++g) {
                int row = m0 + g + 8 * half;
                float v = acc[t][g] + bb;
                h[(size_t)row * H1DIM + nb + m] = v > 0.f ? v : 0.f;
            }
        } else {
#pragma unroll
            for (int g = 0; g < 8; ++g) {
                int row = m0 + g + 8 * half;
                if (row < NNODES) {
                    float v = acc[t][g] + bb;
                    h[(size_t)row * H1DIM + nb + m] = v > 0.f ? v : 0.f;
                }
            }
        }
    }
}

// ---- layer-2 scatter: agg2[tgt,:] += (comp2[r]/cnt[tgt,r]) * h[src,:] -----
__global__ void scatter2(const float* __restrict__ h,
                         const int* __restrict__ ei, const int* __restrict__ et,
                         const float* __restrict__ comp2,
                         const int* __restrict__ cnt,
                         float* __restrict__ agg2) {
    long long idx = (long long)blockIdx.x * blockDim.x + threadIdx.x;
    if (idx >= (long long)NEDGE * H1DIM) return;
    int e = (int)(idx >> 6);
    int j = (int)(idx & 63);
    int src = ei[e];
    int tgt = ei[NEDGE + e];
    int r   = et[e];
    int c   = cnt[tgt * NREL + r];
    float coef = comp2[r] / (float)(c > 1 ? c : 1);
    unsafeAtomicAdd(&agg2[(size_t)tgt * H1DIM + j],
                    coef * h[(size_t)src * H1DIM + j]);
}

// ---- layer-2 dense (K=64, OUT=4) + row softmax ----------------------------
__global__ void layer2_softmax(const float* __restrict__ h,
                               const float* __restrict__ agg2,
                               const float* __restrict__ B2,    /* [64,4] */
                               const float* __restrict__ root2, /* [64,4] */
                               const float* __restrict__ bias2, /* [4] */
                               float* __restrict__ z) {
    int n = blockIdx.x * blockDim.x + threadIdx.x;
    if (n >= NNODES) return;
    float o[OUTDIM] = { bias2[0], bias2[1], bias2[2], bias2[3] };
    const float* hr = h    + (size_t)n * H1DIM;
    const float* ar = agg2 + (size_t)n * H1DIM;
    for (int k = 0; k < H1DIM; ++k) {
        float hv = hr[k], av = ar[k];
#pragma unroll
        for (int j = 0; j < OUTDIM; ++j)
            o[j] += av * B2[k * OUTDIM + j] + hv * root2[k * OUTDIM + j];
    }
    float mx = fmaxf(fmaxf(o[0], o[1]), fmaxf(o[2], o[3]));
    float s = 0.f;
#pragma unroll
    for (int j = 0; j < OUTDIM; ++j) { o[j] = expf(o[j] - mx); s += o[j]; }
    float inv = 1.f / s;
#pragma unroll
    for (int j = 0; j < OUTDIM; ++j) z[(size_t)n * OUTDIM + j] = o[j] * inv;
}

// ---- bilinear decode + sigmoid over concat(pos, neg) edges ----------------
__global__ void decode(const float* __restrict__ z,
                       const int* __restrict__ pos, const int* __restrict__ neg,
                       const float* __restrict__ W, /* [4,4] */
                       float* __restrict__ out) {
    int i = blockIdx.x * blockDim.x + threadIdx.x;
    if (i >= 2 * EDEC) return;
    int a, b;
    if (i < EDEC) { a = pos[i];        b = pos[EDEC + i]; }
    else          { int j = i - EDEC; a = neg[j]; b = neg[EDEC + j]; }
    const float* za = z + (size_t)a * OUTDIM;
    const float* zb = z + (size_t)b * OUTDIM;
    float l = 0.f;
#pragma unroll
    for (int p = 0; p < OUTDIM; ++p)
#pragma unroll
        for (int q = 0; q < OUTDIM; ++q)
            l += za[p] * W[p * OUTDIM + q] * zb[q];
    out[i] = 1.f / (1.f + expf(-l));
}

extern "C" void kernel_launch(void* const* d_in, const int* in_sizes, int n_in,
                              void* d_out, int out_size, void* d_ws, size_t ws_size,
                              hipStream_t stream) {
    const float* x      = (const float*)d_in[0];
    const float* comp1  = (const float*)d_in[1];
    const float* bases1 = (const float*)d_in[2];   // [1,3000,64] -> [3000,64]
    const float* root1  = (const float*)d_in[3];
    const float* bias1  = (const float*)d_in[4];
    const float* comp2  = (const float*)d_in[5];
    const float* bases2 = (const float*)d_in[6];   // [1,64,4] -> [64,4]
    const float* root2  = (const float*)d_in[7];
    const float* bias2  = (const float*)d_in[8];
    const float* bil_w  = (const float*)d_in[9];   // [1,4,4]
    const int* edge_index = (const int*)d_in[10];  // [2,E]
    const int* edge_type  = (const int*)d_in[11];  // [E]
    const int* pos_ei     = (const int*)d_in[12];  // [2,ED]
    const int* neg_ei     = (const int*)d_in[13];  // [2,ED]
    float* out = (float*)d_out;

    // workspace carve-out (256B aligned)
    char* ws = (char*)d_ws;
    size_t off = 0;
    auto carve = [&](size_t bytes) {
        void* p = ws + off;
        off = (off + bytes + 255) & ~(size_t)255;
        return p;
    };
    int*   cnt  = (int*)  carve((size_t)NNODES * NREL  * sizeof(int));
    float* agg1 = (float*)carve((size_t)NNODES * INFEAT * sizeof(float));
    float* h    = (float*)carve((size_t)NNODES * H1DIM * sizeof(float));
    float* agg2 = (float*)carve((size_t)NNODES * H1DIM * sizeof(float));
    float* z    = (float*)carve((size_t)NNODES * OUTDIM * sizeof(float));

    hipMemsetAsync(cnt,  0, (size_t)NNODES * NREL  * sizeof(int),   stream);
    hipMemsetAsync(agg1, 0, (size_t)NNODES * INFEAT * sizeof(float), stream);
    hipMemsetAsync(agg2, 0, (size_t)NNODES * H1DIM * sizeof(float), stream);

    count_edges<<<(NEDGE + 255) / 256, 256, 0, stream>>>(edge_index, edge_type, cnt);
    scatter1<<<NEDGE, 256, 0, stream>>>(x, edge_index, edge_type, comp1, cnt, agg1);
    gemm1_wmma_relu<<<(NNODES + 15) / 16, 32, 0, stream>>>(
        agg1, x, bases1, root1, bias1, h);
    scatter2<<<((long long)NEDGE * H1DIM + 255) / 256, 256, 0, stream>>>(
        h, edge_index, edge_type, comp2, cnt, agg2);
    layer2_softmax<<<(NNODES + 255) / 256, 256, 0, stream>>>(
        h, agg2, bases2, root2, bias2, z);
    decode<<<(2 * EDEC + 255) / 256, 256, 0, stream>>>(z, pos_ei, neg_ei, bil_w, out);
}